// local_seg_model_5042291605823
// MI455X (gfx1250) — compile-verified
//
#include <hip/hip_runtime.h>

typedef __attribute__((ext_vector_type(2))) float v2f;
typedef __attribute__((ext_vector_type(8))) float v8f;

#define J_TOTAL 65536   // B*N
#define NPTS    4096
#define NBATCH  16

// ---------------------------------------------------------------- x2 = |p|^2
__global__ void k_x2(const float* __restrict__ pts, float* __restrict__ x2) {
  int j = blockIdx.x * 256 + threadIdx.x;
  float a = pts[j * 3 + 0], b = pts[j * 3 + 1], c = pts[j * 3 + 2];
  x2[j] = a * a + b * b + c * c;
}

// ---------------------------------------------------------------- KNN (K=8)
// one wave32 per query point; per-batch points + x2 staged in LDS (64 KB)
__global__ __launch_bounds__(256) void k_knn(const float* __restrict__ pts,
                                             const float* __restrict__ x2g,
                                             int* __restrict__ idxo) {
  __shared__ float sp[NPTS * 3];
  __shared__ float sx[NPTS];
  int b = blockIdx.x >> 9;                 // 512 blocks per batch
  int nbase = (blockIdx.x & 511) * 8;
  const float* pbase = pts + (size_t)b * NPTS * 3;
  for (int i = threadIdx.x; i < NPTS * 3; i += 256) sp[i] = pbase[i];
  for (int i = threadIdx.x; i < NPTS; i += 256) sx[i] = x2g[b * NPTS + i];
  __syncthreads();

  int wave = threadIdx.x >> 5, lane = threadIdx.x & 31;
  int n = nbase + wave;
  float qx = sp[n * 3 + 0], qy = sp[n * 3 + 1], qz = sp[n * 3 + 2];
  float q2 = sx[n];

  float bd[8]; int bi[8];
#pragma unroll
  for (int s = 0; s < 8; ++s) { bd[s] = 3.4e38f; bi[s] = 0x7fffffff; }

  for (int m = lane; m < NPTS; m += 32) {
    float px = sp[m * 3], py = sp[m * 3 + 1], pz = sp[m * 3 + 2];
    float d = q2 + sx[m] - 2.f * (qx * px + qy * py + qz * pz);
    if (d < bd[7] || (d == bd[7] && m < bi[7])) {
      bd[7] = d; bi[7] = m;
#pragma unroll
      for (int s = 7; s > 0; --s) {
        if ((bd[s] < bd[s - 1]) || (bd[s] == bd[s - 1] && bi[s] < bi[s - 1])) {
          float tf = bd[s]; bd[s] = bd[s - 1]; bd[s - 1] = tf;
          int ti = bi[s]; bi[s] = bi[s - 1]; bi[s - 1] = ti;
        }
      }
    }
  }
  // 8 wave-wide min-selections; tie-break on smaller index (top_k stability)
  int* orow = idxo + ((size_t)(b * NPTS + n)) * 8;
#pragma unroll
  for (int sel = 0; sel < 8; ++sel) {
    float d = bd[0]; int i = bi[0];
#pragma unroll
    for (int off = 16; off > 0; off >>= 1) {
      float od = __shfl_xor(d, off, 32);
      int oi = __shfl_xor(i, off, 32);
      if (od < d || (od == d && oi < i)) { d = od; i = oi; }
    }
    if (bi[0] == i) {                       // winner lane pops its head
#pragma unroll
      for (int s = 0; s < 7; ++s) { bd[s] = bd[s + 1]; bi[s] = bi[s + 1]; }
      bd[7] = 3.4e38f; bi[7] = 0x7fffffff;
    }
    if (lane == 0) orow[sel] = i;
  }
}

// ---------------------------------------------------------------- layer1 (64x3)
__global__ void k_l1(const float* __restrict__ pts, const float* __restrict__ w1,
                     const float* __restrict__ b1, float* __restrict__ y) {
  int t = blockIdx.x * 256 + threadIdx.x;
  int c = t >> 16, j = t & 65535;
  float v = b1[c];
  v = fmaf(w1[c * 3 + 0], pts[j * 3 + 0], v);
  v = fmaf(w1[c * 3 + 1], pts[j * 3 + 1], v);
  v = fmaf(w1[c * 3 + 2], pts[j * 3 + 2], v);
  y[(size_t)c * J_TOTAL + j] = v;
}

// ---------------------------------------------------------------- BN stats
// one block per channel; deterministic fixed-order reduction (double acc)
__global__ __launch_bounds__(256) void k_stats(const float* __restrict__ Y,
                                               const float* __restrict__ g,
                                               const float* __restrict__ be,
                                               float* __restrict__ scale,
                                               float* __restrict__ shift) {
  __shared__ double ss[256], sq[256];
  int c = blockIdx.x;
  const float* row = Y + (size_t)c * J_TOTAL;
  double s = 0.0, q = 0.0;
  for (int j = threadIdx.x; j < J_TOTAL; j += 256) {
    double v = (double)row[j]; s += v; q += v * v;
  }
  ss[threadIdx.x] = s; sq[threadIdx.x] = q; __syncthreads();
  for (int o = 128; o > 0; o >>= 1) {
    if (threadIdx.x < o) { ss[threadIdx.x] += ss[threadIdx.x + o]; sq[threadIdx.x] += sq[threadIdx.x + o]; }
    __syncthreads();
  }
  if (threadIdx.x == 0) {
    float m = (float)(ss[0] / (double)J_TOTAL);
    float v = (float)(sq[0] / (double)J_TOTAL) - m * m;
    float sc = g[c] * rsqrtf(v + 1e-5f);
    scale[c] = sc; shift[c] = be[c] - m * sc;
  }
}

// ---------------------------------------------------------------- WMMA GEMM
// Y(O,J) = W(O,C) x relu(xs*X+xh)(C,J) + epilogue
// block = 8 waves, tile 128(M) x 64(J); fp32 WMMA 16x16x4.
// Activation tile stored TRANSPOSED in LDS: xt[n][k], row stride 68 dwords
// -> B fragment is one aligned ds_load_b64 (contiguous v2f, no movs),
//    and the 32-lane read pattern covers all 64 banks (4n+k0+{0..3}).
// Next K-tile is prefetched into registers to overlap global latency.
// EPI 0: +bias[o]   EPI 1: +srow[o]*lf[j] + gvec[b][o]  (H1 rank-1/broadcast)
template <int EPI>
__global__ __launch_bounds__(256) void k_gemm(
    const float* __restrict__ W, int wld, int wcol0, int C,
    const float* __restrict__ X, const float* __restrict__ xs, const float* __restrict__ xh,
    const float* __restrict__ bias,
    const float* __restrict__ srow, const float* __restrict__ lf,
    const float* __restrict__ gvec, float* __restrict__ Y) {
  __shared__ float xt[64][68];                 // [j-in-tile][k], padded stride 68
  int wave = threadIdx.x >> 5, lane = threadIdx.x & 31;
  int m16 = lane & 15, half = lane >> 4;
  int ob = blockIdx.y * 128 + wave * 16;
  int jb = blockIdx.x * 64;
  v8f acc0 = {}, acc1 = {}, acc2 = {}, acc3 = {};

  // per-thread fill slot: jj fastest for coalesced global reads
  int fjj = threadIdx.x & 63;
  int fc0 = threadIdx.x >> 6;                  // 4 rows per pass, 16 passes

  float pv[16];
#pragma unroll
  for (int it = 0; it < 16; ++it)              // prefetch K-tile 0
    pv[it] = X[(size_t)(fc0 + it * 4) * J_TOTAL + jb + fjj];

  for (int kt = 0; kt < C; kt += 64) {
    __syncthreads();                           // protect LDS from prior compute
#pragma unroll
    for (int it = 0; it < 16; ++it) {          // transform + transposed store
      int c = fc0 + it * 4;
      float v = pv[it];
      if (xs) v = fmaxf(fmaf(xs[kt + c], v, xh[kt + c]), 0.f);
      xt[fjj][c] = v;
    }
    __syncthreads();
    if (kt + 64 < C) {                         // prefetch next K-tile
#pragma unroll
      for (int it = 0; it < 16; ++it)
        pv[it] = X[(size_t)(kt + 64 + fc0 + it * 4) * J_TOTAL + jb + fjj];
    }
#pragma unroll
    for (int k0 = 0; k0 < 64; k0 += 4) {
      int kk = kt + k0 + 2 * half;
      const float* wr = W + (size_t)(ob + m16) * wld + wcol0 + kk;
      v2f a; a.x = wr[0]; a.y = wr[1];         // A(m,k): lane=m+16*(k>>1), vgpr=k&1
      int kr = k0 + 2 * half;
      v2f b0 = *reinterpret_cast<const v2f*>(&xt[m16][kr]);
      v2f b1 = *reinterpret_cast<const v2f*>(&xt[16 + m16][kr]);
      v2f b2 = *reinterpret_cast<const v2f*>(&xt[32 + m16][kr]);
      v2f b3 = *reinterpret_cast<const v2f*>(&xt[48 + m16][kr]);
      acc0 = __builtin_amdgcn_wmma_f32_16x16x4_f32(false, a, false, b0, (short)0, acc0, false, false);
      acc1 = __builtin_amdgcn_wmma_f32_16x16x4_f32(false, a, false, b1, (short)0, acc1, false, false);
      acc2 = __builtin_amdgcn_wmma_f32_16x16x4_f32(false, a, false, b2, (short)0, acc2, false, false);
      acc3 = __builtin_amdgcn_wmma_f32_16x16x4_f32(false, a, false, b3, (short)0, acc3, false, false);
    }
  }
#pragma unroll
  for (int nf = 0; nf < 4; ++nf) {
    v8f acc = nf == 0 ? acc0 : nf == 1 ? acc1 : nf == 2 ? acc2 : acc3;
    int j = jb + nf * 16 + m16;
#pragma unroll
    for (int r = 0; r < 8; ++r) {              // D(m,n): vgpr r -> m=r+8*half
      int o = ob + r + 8 * half;
      float v = acc[r];
      if (EPI == 0) v += bias[o];
      else          v += srow[o] * lf[j] + gvec[((j >> 12) << 9) + o];
      Y[(size_t)o * J_TOTAL + j] = v;
    }
  }
}

// ------------------------------------------------- global max per (b,channel)
__global__ __launch_bounds__(256) void k_gmax(const float* __restrict__ X,
                                              const float* __restrict__ sc,
                                              const float* __restrict__ sh,
                                              float* __restrict__ gmax) {
  __shared__ float red[256];
  int c = blockIdx.x & 511, b = blockIdx.x >> 9;
  const float* row = X + (size_t)c * J_TOTAL + b * NPTS;
  float s = sc[c], h = sh[c], mx = -3.4e38f;
  for (int n = threadIdx.x; n < NPTS; n += 256)
    mx = fmaxf(mx, fmaxf(fmaf(s, row[n], h), 0.f));
  red[threadIdx.x] = mx; __syncthreads();
  for (int o = 128; o > 0; o >>= 1) {
    if (threadIdx.x < o) red[threadIdx.x] = fmaxf(red[threadIdx.x], red[threadIdx.x + o]);
    __syncthreads();
  }
  if (threadIdx.x == 0) gmax[b * 512 + c] = red[0];
}

// ------------------------------- local feature scalar (reference gather bug)
// lf[b,n] = max_k relu(bn(x))[b, k, idx[b,n,k]]   (only channels 0..7 used)
__global__ void k_lf(const float* __restrict__ X, const float* __restrict__ sc,
                     const float* __restrict__ sh, const int* __restrict__ idx,
                     float* __restrict__ lf) {
  int j = blockIdx.x * 256 + threadIdx.x;
  int b = j >> 12;
  const int* row = idx + (size_t)j * 8;
  float mx = -3.4e38f;
#pragma unroll
  for (int k = 0; k < 8; ++k) {
    int mi = row[k];
    float v = fmaxf(fmaf(sc[k], X[(size_t)k * J_TOTAL + (b << 12) + mi], sh[k]), 0.f);
    mx = fmaxf(mx, v);
  }
  lf[j] = mx;
}

// ------------------------------------------- H1 rank-1 / per-batch prep terms
__global__ void k_srow(const float* __restrict__ hw1, float* __restrict__ srow) {
  int o = blockIdx.x * 256 + threadIdx.x;      // 512 threads total
  float s = 0.f;
  for (int c = 0; c < 512; ++c) s += hw1[o * 1088 + c];
  srow[o] = s;
}

__global__ void k_gvec(const float* __restrict__ hw1, const float* __restrict__ hb1,
                       const float* __restrict__ gmax, float* __restrict__ gvec) {
  int t = blockIdx.x * 256 + threadIdx.x;      // 16*512
  int b = t >> 9, o = t & 511;
  float s = hb1[o];
  const float* gm = gmax + b * 512;
  for (int c = 0; c < 512; ++c) s = fmaf(hw1[o * 1088 + 576 + c], gm[c], s);
  gvec[t] = s;
}

// ---------------------------------------------------------------- head (6x128)
__global__ void k_h4(const float* __restrict__ X, const float* __restrict__ sc,
                     const float* __restrict__ sh, const float* __restrict__ w,
                     const float* __restrict__ bias, float* __restrict__ out) {
  int j = blockIdx.x * 256 + threadIdx.x;
  float acc[6];
#pragma unroll
  for (int o = 0; o < 6; ++o) acc[o] = bias[o];
  for (int c = 0; c < 128; ++c) {
    float v = fmaxf(fmaf(sc[c], X[(size_t)c * J_TOTAL + j], sh[c]), 0.f);
#pragma unroll
    for (int o = 0; o < 6; ++o) acc[o] = fmaf(w[o * 128 + c], v, acc[o]);
  }
#pragma unroll
  for (int o = 0; o < 6; ++o) out[(size_t)j * 6 + o] = acc[o];
}

// ================================================================ launcher
extern "C" void kernel_launch(void* const* d_in, const int* in_sizes, int n_in,
                              void* d_out, int out_size, void* d_ws, size_t ws_size,
                              hipStream_t stream) {
  const float* points = (const float*)d_in[0];
  const float* w1  = (const float*)d_in[1];  const float* b1  = (const float*)d_in[2];
  const float* g1  = (const float*)d_in[3];  const float* be1 = (const float*)d_in[4];
  const float* w3  = (const float*)d_in[5];  const float* b3  = (const float*)d_in[6];
  const float* g3  = (const float*)d_in[7];  const float* be3 = (const float*)d_in[8];
  const float* hw1 = (const float*)d_in[9];  const float* hb1 = (const float*)d_in[10];
  const float* hg1 = (const float*)d_in[11]; const float* hbe1= (const float*)d_in[12];
  const float* hw2 = (const float*)d_in[13]; const float* hb2 = (const float*)d_in[14];
  const float* hg2 = (const float*)d_in[15]; const float* hbe2= (const float*)d_in[16];
  const float* hw3 = (const float*)d_in[17]; const float* hb3 = (const float*)d_in[18];
  const float* hg3 = (const float*)d_in[19]; const float* hbe3= (const float*)d_in[20];
  const float* hw4 = (const float*)d_in[21]; const float* hb4 = (const float*)d_in[22];

  const size_t J = J_TOTAL;
  float* ws = (float*)d_ws;
  size_t off = 0;
  float* x2   = ws + off; off += J;
  float* lf   = ws + off; off += J;
  int*   idx  = (int*)(ws + off); off += J * 8;
  float* sc1  = ws + off; off += 64;   float* sh1  = ws + off; off += 64;
  float* sc3  = ws + off; off += 512;  float* sh3  = ws + off; off += 512;
  float* sch1 = ws + off; off += 512;  float* shh1 = ws + off; off += 512;
  float* sch2 = ws + off; off += 256;  float* shh2 = ws + off; off += 256;
  float* sch3 = ws + off; off += 128;  float* shh3 = ws + off; off += 128;
  float* gmax = ws + off; off += 16 * 512;
  float* srow = ws + off; off += 512;
  float* gvec = ws + off; off += 16 * 512;
  float* y1   = ws + off; off += 64 * J;     // layer-1 pre-activation
  float* x3   = ws + off; off += 512 * J;    // layer-3 pre-act; reused for h2pre
  float* bufA = ws + off; off += 512 * J;    // h1pre; reused for h3pre
  float* h1 = bufA;
  float* h2 = x3;                            // x3 dead after gmax/lf
  float* h3 = bufA;                          // h1 dead after H2 GEMM

  k_x2 <<<J / 256, 256, 0, stream>>>(points, x2);
  k_knn<<<NBATCH * (NPTS / 8), 256, 0, stream>>>(points, x2, idx);
  k_l1 <<<64 * J / 256, 256, 0, stream>>>(points, w1, b1, y1);
  k_stats<<<64, 256, 0, stream>>>(y1, g1, be1, sc1, sh1);

  k_gemm<0><<<dim3(J / 64, 4), 256, 0, stream>>>(w3, 64, 0, 64,
      y1, sc1, sh1, b3, nullptr, nullptr, nullptr, x3);
  k_stats<<<512, 256, 0, stream>>>(x3, g3, be3, sc3, sh3);

  k_gmax<<<NBATCH * 512, 256, 0, stream>>>(x3, sc3, sh3, gmax);
  k_lf  <<<J / 256, 256, 0, stream>>>(x3, sc3, sh3, idx, lf);
  k_srow<<<2, 256, 0, stream>>>(hw1, srow);
  k_gvec<<<NBATCH * 512 / 256, 256, 0, stream>>>(hw1, hb1, gmax, gvec);

  k_gemm<1><<<dim3(J / 64, 4), 256, 0, stream>>>(hw1, 1088, 512, 64,
      y1, sc1, sh1, nullptr, srow, lf, gvec, h1);
  k_stats<<<512, 256, 0, stream>>>(h1, hg1, hbe1, sch1, shh1);

  k_gemm<0><<<dim3(J / 64, 2), 256, 0, stream>>>(hw2, 512, 0, 512,
      h1, sch1, shh1, hb2, nullptr, nullptr, nullptr, h2);
  k_stats<<<256, 256, 0, stream>>>(h2, hg2, hbe2, sch2, shh2);

  k_gemm<0><<<dim3(J / 64, 1), 256, 0, stream>>>(hw3, 256, 0, 256,
      h2, sch2, shh2, hb3, nullptr, nullptr, nullptr, h3);
  k_stats<<<128, 256, 0, stream>>>(h3, hg3, hbe3, sch3, shh3);

  k_h4<<<J / 256, 256, 0, stream>>>(h3, sch3, shh3, hw4, hb4, (float*)d_out);
}